// GenericParamEncoding_79645873537411
// MI455X (gfx1250) — compile-verified
//
#include <hip/hip_runtime.h>

typedef __attribute__((ext_vector_type(2))) float v2f;
typedef __attribute__((ext_vector_type(8))) float v8f;

#define NDIM 512

// out[b,i,p] = W[i,b] * params[p,b]
// One block: one b, 128 rows of i (8 waves x 16 rows), full p sweep.
// Each wave computes 16x16 output tiles via V_WMMA_F32_16X16X4_F32 as a
// rank-1 outer product. Only A's K=1..3 slots must be true zeros; B's
// K=1..3 slots may hold arbitrary finite data (multiplied by 0), which
// lets us load B from LDS unconditionally (no EXEC-masked branches).
__global__ __launch_bounds__(256) void outer_wmma_kernel(
    const float* __restrict__ params,  // [p, b] row-major, 512x512
    const float* __restrict__ W,       // [i, b] row-major, 512x512
    float* __restrict__ out)           // [b, i, p], 512^3
{
    const int b      = blockIdx.x;       // 0..511
    const int ichunk = blockIdx.y;       // 0..3 -> 128 rows of i each
    const int t      = threadIdx.x;      // 0..255
    const int wave   = t >> 5;           // 0..7
    const int lane   = t & 31;

    __shared__ float sP[NDIM];           // params[:, b]
    __shared__ float sW[128];            // W[ichunk*128 .. +128, b]

    // Stage the two strided columns into LDS (one pass; inputs are tiny).
    sP[t]       = params[(size_t)t * NDIM + b];
    sP[t + 256] = params[(size_t)(t + 256) * NDIM + b];
    if (t < 128) sW[t] = W[(size_t)(ichunk * 128 + t) * NDIM + b];
    __syncthreads();

    const int  i0 = ichunk * 128 + wave * 16;
    const bool lo = (lane < 16);
    const int  n  = lane & 15;

    // A matrix 16x4 f32: VGPR0 lanes 0-15 = (M=lane, K=0); lanes 16-31 = K=2
    // (must be 0); VGPR1 = K=1/K=3 (must be 0). Unconditional LDS read, then
    // a register select (no branch).
    const float wv = sW[wave * 16 + n];
    v2f a;
    a.x = lo ? wv : 0.0f;
    a.y = 0.0f;

    // D layout: VGPR r, lanes 0-15 -> (M=r, N=lane); lanes 16-31 -> (M=8+r, N=lane-16)
    float* outrow = out + (size_t)b * NDIM * NDIM
                        + (size_t)(i0 + (lo ? 0 : 8)) * NDIM + n;

    for (int p0 = 0; p0 < NDIM; p0 += 16) {
        // B matrix 4x16 f32: VGPR0 lanes 0-15 = (K=0, N=lane). Lanes 16-31
        // land in the K=2 row, which is multiplied by A's zeros -> harmless.
        v2f bv;
        bv.x = sP[p0 + n];   // unconditional: no EXEC masking
        bv.y = 0.0f;

        v8f c = {};
        // (neg_a, A, neg_b, B, c_mod, C, reuse_a, reuse_b)
        c = __builtin_amdgcn_wmma_f32_16x16x4_f32(
            false, a, false, bv, (short)0, c, false, false);

#pragma unroll
        for (int r = 0; r < 8; ++r) {
            // lanes 0-15 write rows i0+r, lanes 16-31 rows i0+8+r; 16-lane
            // contiguous 64B segments per row, merged to full lines in L2.
            outrow[(size_t)r * NDIM + p0] = c[r];
        }
    }
}

extern "C" void kernel_launch(void* const* d_in, const int* in_sizes, int n_in,
                              void* d_out, int out_size, void* d_ws, size_t ws_size,
                              hipStream_t stream) {
    const float* params = (const float*)d_in[0];  // [n_params, batch]
    const float* W      = (const float*)d_in[1];  // [P, P]
    float*       out    = (float*)d_out;          // [batch, P, n_params]

    dim3 grid(NDIM, 4, 1);   // 512 b-slabs x 4 i-chunks
    dim3 block(256, 1, 1);   // 8 waves
    outer_wmma_kernel<<<grid, block, 0, stream>>>(params, W, out);
}